// PreBasicLinear_module_49752901157256
// MI455X (gfx1250) — compile-verified
//
#include <hip/hip_runtime.h>

// Fused: h = X @ W^T + b ; row-wise LayerNorm(128) ; LeakyReLU(0.2)
// X:[N,128] f32, W:[128,128] f32, out:[N,128] f32.
// HBM-bound: mandatory traffic = 512MB (X) + 512MB (out) ~= 44us @ 23.3TB/s.
// GEMM uses V_WMMA_F32_16X16X4_F32 (full fp32 precision, compute is not the
// bottleneck). W is DMA'd once per block into LDS by the Tensor Data Mover
// with pad_enable (pitch 128->132 floats) so all B-fragment ds_load_b64
// reads are LDS-bank-conflict-free; the vmem pipe then only streams X/out.

typedef float v2f __attribute__((ext_vector_type(2)));
typedef float v8f __attribute__((ext_vector_type(8)));
typedef unsigned int u32x4 __attribute__((ext_vector_type(4)));
typedef int i32x4 __attribute__((ext_vector_type(4)));
typedef int i32x8 __attribute__((ext_vector_type(8)));

#define TILES_PER_BLOCK 4                  // 4 waves/block, one 16-row tile each
#define THREADS (TILES_PER_BLOCK * 32)     // wave32
#define WPITCH 132                         // padded LDS pitch (floats): 4-bank lane stride

#if __has_builtin(__builtin_amdgcn_tensor_load_to_lds) && \
    __has_builtin(__builtin_amdgcn_s_wait_tensorcnt)
#define HAVE_TDM 1
#else
#define HAVE_TDM 0
#endif

__global__ __launch_bounds__(THREADS)
void fused_linear_ln_lrelu(const float* __restrict__ X,
                           const float* __restrict__ Wt,     // [128,128] row-major
                           const float* __restrict__ bias,   // [128]
                           const float* __restrict__ gamma,  // [128]
                           const float* __restrict__ beta,   // [128]
                           float* __restrict__ out,          // [N,128]
                           int numTiles)
{
    // Padded W copy (128 x 132 floats = 66 KB); reused post-GEMM as the
    // per-wave 16x132 epilogue staging area (4 x 8.25 KB).
    __shared__ float sW[128 * WPITCH];

    const int lane    = threadIdx.x & 31;
    const int wave    = threadIdx.x >> 5;
    const int rowTile = blockIdx.x * TILES_PER_BLOCK + wave;
    const bool active = rowTile < numTiles;   // wave-uniform => EXEC all-1 for WMMA
    const int row0    = rowTile << 4;

    const int mrow  = lane & 15;   // A-matrix row / W row (== output col)
    const int khalf = lane >> 4;   // which K-pair of the K=4 chunk this lane holds

    // ---- Stage W into LDS (padded pitch 132) --------------------------------
#if HAVE_TDM
    if (wave == 0) {
        // Tensor DMA: 128x128 f32 tile, row-linear in memory, pad 4 dwords
        // after every 128 dwords in LDS (pitch 132).
        unsigned long long ga = (unsigned long long)(uintptr_t)Wt;
        unsigned int ldsaddr  = (unsigned int)(uintptr_t)&sW[0]; // low 32 = LDS offset
        u32x4 g0;
        g0.x = 1u;                                   // count=1, user descriptor
        g0.y = ldsaddr;                              // lds_addr [63:32]
        g0.z = (unsigned int)ga;                     // global_addr low
        g0.w = (unsigned int)(ga >> 32) | (2u << 30);// global_addr hi | type=2
        i32x8 g1;
        g1[0] = (2 << 16)      // data_size = 4 bytes
              | (1 << 20)      // pad_enable
              | (6 << 22)      // pad_interval: every 128 dwords (one row)
              | (3 << 25);     // pad_amount: 4 dwords  -> pitch 132
        g1[1] = (128 << 16);   // tensor_dim0 = 128 (low 16 bits)
        g1[2] = (128 << 16);   // tensor_dim0 hi=0 | tensor_dim1 = 128
        g1[3] = (128 << 16);   // tensor_dim1 hi=0 | tile_dim0 = 128
        g1[4] = 128;           // tile_dim1 = 128, tile_dim2 = 0
        g1[5] = 128;           // tensor_dim0_stride = 128 elements
        g1[6] = (int)((16384u & 0xFFFFu) << 16); // tensor_dim1_stride low16
        g1[7] = 0;
        i32x4 gz4 = {0, 0, 0, 0};                // groups 2/3 unused (2D tensor)
        i32x8 gz8 = {0, 0, 0, 0, 0, 0, 0, 0};    // extra group (6-arg form), unused
        __builtin_amdgcn_tensor_load_to_lds(g0, g1, gz4, gz4, gz8, 0);
        __builtin_amdgcn_s_wait_tensorcnt(0);
    }
#else
    // Fallback: cooperative padded copy, 128 float4 per thread-iteration.
    for (int idx = threadIdx.x; idx < 128 * 32; idx += THREADS) {
        int r = idx >> 5, c4 = (idx & 31) << 2;
        float4 v = *(const float4*)(Wt + r * 128 + c4);
        *(float4*)(&sW[r * WPITCH + c4]) = v;
    }
#endif
    __syncthreads();

    // ---- GEMM: 16x128 tile per wave, K=128 as 32 chunks of 4 ----------------
    v8f acc[8] = {};
    if (active) {
        // A frag: X[(row0+mrow)*128 + 4*kc + 2*khalf]  (one global b64 per chunk)
        const float* xp = X + (long)(row0 + mrow) * 128 + khalf * 2;
        // B frag: sW[(16*t+mrow)*132 + 4*kc + 2*khalf] (conflict-free ds b64)
        const float* wl = sW + mrow * WPITCH + khalf * 2;

#pragma unroll 4
        for (int kc = 0; kc < 32; ++kc) {
            v2f a = *(const v2f*)(xp + kc * 4);
#pragma unroll
            for (int t = 0; t < 8; ++t) {
                v2f bf = *(const v2f*)(wl + t * (16 * WPITCH) + kc * 4);
                acc[t] = __builtin_amdgcn_wmma_f32_16x16x4_f32(
                    false, a, false, bf, (short)0, acc[t], false, false);
            }
        }
    }

    __syncthreads();   // all waves done reading W -> safe to reuse sW

    // ---- Spill C tiles to LDS (per-wave 16x132 region) ----------------------
    float* h = &sW[wave * 16 * WPITCH];
    if (active) {
        const int crow = khalf * 8;  // VGPR v -> row v (lanes 0-15) or v+8
#pragma unroll
        for (int t = 0; t < 8; ++t)
#pragma unroll
            for (int v = 0; v < 8; ++v)
                h[(crow + v) * WPITCH + t * 16 + mrow] = acc[t][v];
    }

    __syncthreads();

    // ---- Epilogue: bias + row LayerNorm(128) + LeakyReLU(0.2) ---------------
    if (active) {
        const int c4 = lane << 2;   // 4 consecutive features per lane
        float4 ga = *(const float4*)(gamma + c4);
        float4 be = *(const float4*)(beta  + c4);
        float4 bi = *(const float4*)(bias  + c4);
        float* op = out + (long)row0 * 128 + c4;

#pragma unroll 4
        for (int r = 0; r < 16; ++r) {
            float4 v = *(const float4*)(h + r * WPITCH + c4);
            v.x += bi.x; v.y += bi.y; v.z += bi.z; v.w += bi.w;

            float s = v.x + v.y + v.z + v.w;
            float q = v.x * v.x + v.y * v.y + v.z * v.z + v.w * v.w;
#pragma unroll
            for (int off = 16; off; off >>= 1) {       // wave32 reduction
                s += __shfl_xor(s, off, 32);
                q += __shfl_xor(q, off, 32);
            }
            float mean = s * (1.0f / 128.0f);
            float var  = q * (1.0f / 128.0f) - mean * mean;
            float inv  = rsqrtf(var + 1e-5f);

            float4 o;
            o.x = (v.x - mean) * inv * ga.x + be.x;
            o.y = (v.y - mean) * inv * ga.y + be.y;
            o.z = (v.z - mean) * inv * ga.z + be.z;
            o.w = (v.w - mean) * inv * ga.w + be.w;
            o.x = o.x >= 0.f ? o.x : 0.2f * o.x;
            o.y = o.y >= 0.f ? o.y : 0.2f * o.y;
            o.z = o.z >= 0.f ? o.z : 0.2f * o.z;
            o.w = o.w >= 0.f ? o.w : 0.2f * o.w;
            *(float4*)(op + r * 128) = o;              // 512B/row coalesced
        }
    }
}

extern "C" void kernel_launch(void* const* d_in, const int* in_sizes, int n_in,
                              void* d_out, int out_size, void* d_ws, size_t ws_size,
                              hipStream_t stream) {
    // setup_inputs order: input_x, batch(int64, unused), W, b, gamma, beta
    const float* X     = (const float*)d_in[0];
    const float* W     = (const float*)d_in[2];
    const float* bias  = (const float*)d_in[3];
    const float* gamma = (const float*)d_in[4];
    const float* beta  = (const float*)d_in[5];
    float* out = (float*)d_out;

    const int n        = in_sizes[0] / 128;                 // rows (1,000,000)
    const int numTiles = (n + 15) / 16;                     // 16-row tiles
    const int blocks   = (numTiles + TILES_PER_BLOCK - 1) / TILES_PER_BLOCK;

    fused_linear_ln_lrelu<<<blocks, THREADS, 0, stream>>>(
        X, W, bias, gamma, beta, out, numTiles);
}